// GCN_36455682408682
// MI455X (gfx1250) — compile-verified
//
#include <hip/hip_runtime.h>

typedef float v2f __attribute__((ext_vector_type(2)));
typedef float v8f __attribute__((ext_vector_type(8)));

#define N_NODES   100000
#define N_EDGES   1600000
#define IN_FEATS  256
#define H_FEATS   128
#define N_CLASSES 32

// ---------------- degree accumulation ----------------
__global__ void deg_kernel(const int* __restrict__ src, const int* __restrict__ dst,
                           float* __restrict__ dout, float* __restrict__ din, int nE) {
  int e = blockIdx.x * blockDim.x + threadIdx.x;
  if (e < nE) {
    atomicAdd(&dout[src[e]], 1.0f);
    atomicAdd(&din[dst[e]], 1.0f);
  }
}

// d <- rsqrt(max(d,1)) for both norm arrays
__global__ void invsqrt_kernel(float* __restrict__ a, float* __restrict__ b, int n) {
  int i = blockIdx.x * blockDim.x + threadIdx.x;
  if (i < n) {
    a[i] = rsqrtf(fmaxf(a[i], 1.0f));
    b[i] = rsqrtf(fmaxf(b[i], 1.0f));
  }
}

// ---------------- pack W (KxN row-major) into pair-major float2 ----------------
// Wp[p*N + n] = (W[2p][n], W[2p+1][n]); one b64 load then feeds a B fragment.
__global__ void pack_w_kernel(const float* __restrict__ W, float2* __restrict__ Wp,
                              int K, int N, int nShift) {
  int t = blockIdx.x * blockDim.x + threadIdx.x;
  int total = (K >> 1) * N;
  if (t >= total) return;
  int p = t >> nShift;
  int n = t & (N - 1);
  float2 v;
  v.x = W[(size_t)(2 * p) * N + n];
  v.y = W[(size_t)(2 * p + 1) * N + n];
  Wp[t] = v;
}

// ---------------- fp32 WMMA GEMM: C = A(MxK) @ Bp, row-major ----------------
// blockIdx.x = M tile (16 rows), wave-in-block = N tile. Block stages the
// contiguous 16xK A tile in LDS (rows padded to K+4 dwords: the 32-lane
// fragment read touches all 64 banks exactly once). Inner loop per K-step:
// ds_load_b64 (A frag) + global_load_b64 (packed B frag) + v_wmma.
// A frag (16x4 f32): lane = half*16 + m; v0=A[m][k+2h], v1=A[m][k+2h+1]
// B frag (4x16 f32): lane = half*16 + n; v0=B[k+2h][n], v1=B[k+2h+1][n]
// C/D (16x16 f32):   VGPR r, lanes 0-15: (M=r,N=lane); lanes 16-31: (M=r+8,N=lane-16)
__global__ void wmma_gemm_f32_lds(const float* __restrict__ A,
                                  const float2* __restrict__ Bp,
                                  float* __restrict__ C,
                                  int K, int N, int kShift) {
  extern __shared__ float lds[];
  const int mt   = blockIdx.x;
  const int nt   = threadIdx.x >> 5;
  const int lane = threadIdx.x & 31;
  const int l16  = lane & 15;
  const int half = lane >> 4;
  const int ldsStride = K + 4;

  // cooperative stage of the 16xK A tile (contiguous in memory)
  {
    const float4* Asrc = (const float4*)(A + (size_t)mt * 16 * K);
    const int nvec = (16 * K) >> 2;            // float4 count
    const int cmask = (K >> 2) - 1;            // float4s per row - 1
    for (int i = threadIdx.x; i < nvec; i += blockDim.x) {
      const int r = i >> (kShift - 2);
      const int c = (i & cmask) << 2;
      *(float4*)&lds[r * ldsStride + c] = Asrc[i];
    }
  }
  __syncthreads();

  const float*  Arow = &lds[l16 * ldsStride + half * 2];
  const float2* Bf   = Bp + (size_t)half * N + nt * 16 + l16;  // pair index half, col n

  v8f acc = {};
  for (int k = 0; k < K; k += 4) {
    v2f a, b;
    a.x = Arow[0];
    a.y = Arow[1];
    const float2 bb = *Bf;
    b.x = bb.x;
    b.y = bb.y;
    acc = __builtin_amdgcn_wmma_f32_16x16x4_f32(false, a, false, b, (short)0, acc,
                                                false, false);
    Arow += 4;                // next K-quad in LDS
    Bf += (size_t)2 * N;      // +2 K-pairs
  }

  float* Cp = C + (size_t)(mt * 16 + half * 8) * N + nt * 16 + l16;
#pragma unroll
  for (int r = 0; r < 8; ++r)
    Cp[(size_t)r * N] = acc[r];
}

// ---------------- edge scatter-add with fused source scaling ----------------
// AGG[dst[e], :] += Y[src[e], :] * sout[src[e]]
// cpeShift = log2(F/4); consecutive threads = consecutive 16B chunks of one edge row.
__global__ void scatter_kernel(const float* __restrict__ Y, const int* __restrict__ src,
                               const int* __restrict__ dst, const float* __restrict__ sout,
                               float* __restrict__ AGG, int nE, int F, int cpeShift) {
  long t = (long)blockIdx.x * blockDim.x + threadIdx.x;
  const int cpe = 1 << cpeShift;
  const long total = (long)nE << cpeShift;
  if (t >= total) return;
  const int e = (int)(t >> cpeShift);
  const int c = ((int)t & (cpe - 1)) << 2;
  const int s = src[e];
  const int d = dst[e];
  const float sc = sout[s];
  const float4 v = *(const float4*)(Y + (size_t)s * F + c);
  float* o = AGG + (size_t)d * F + c;
  atomicAdd(o + 0, v.x * sc);
  atomicAdd(o + 1, v.y * sc);
  atomicAdd(o + 2, v.z * sc);
  atomicAdd(o + 3, v.w * sc);
}

// ---------------- dest scale + bias (+ optional ReLU), in place ----------------
__global__ void post_kernel(float* __restrict__ X, const float* __restrict__ sin_,
                            const float* __restrict__ bias, int n, int fShift, int doRelu) {
  long t = (long)blockIdx.x * blockDim.x + threadIdx.x;
  if (t >= ((long)n << fShift)) return;
  const int i = (int)(t >> fShift);
  const int f = (int)t & ((1 << fShift) - 1);
  float v = X[t] * sin_[i] + bias[f];
  if (doRelu) v = fmaxf(v, 0.0f);
  X[t] = v;
}

extern "C" void kernel_launch(void* const* d_in, const int* in_sizes, int n_in,
                              void* d_out, int out_size, void* d_ws, size_t ws_size,
                              hipStream_t stream) {
  (void)in_sizes; (void)n_in; (void)out_size; (void)ws_size;
  const float* x   = (const float*)d_in[0];
  const int*   src = (const int*)d_in[1];
  const int*   dst = (const int*)d_in[2];
  const float* W1  = (const float*)d_in[3];
  const float* b1  = (const float*)d_in[4];
  const float* W2  = (const float*)d_in[5];
  const float* b2  = (const float*)d_in[6];
  float* out = (float*)d_out;

  char* ws = (char*)d_ws;
  size_t off = 0;
  auto take = [&](size_t bytes) {
    void* p = (void*)(ws + off);
    off = (off + bytes + 255) & ~(size_t)255;
    return p;
  };
  float*  s_out = (float*)take((size_t)N_NODES * 4);
  float*  s_in  = (float*)take((size_t)N_NODES * 4);
  float*  Y1    = (float*)take((size_t)N_NODES * H_FEATS * 4);
  float*  AGG1  = (float*)take((size_t)N_NODES * H_FEATS * 4);
  float*  Y2    = (float*)take((size_t)N_NODES * N_CLASSES * 4);
  float2* Wp1   = (float2*)take((size_t)(IN_FEATS / 2) * H_FEATS * 8);
  float2* Wp2   = (float2*)take((size_t)(H_FEATS / 2) * N_CLASSES * 8);

  hipMemsetAsync(s_out, 0, (size_t)N_NODES * 4, stream);
  hipMemsetAsync(s_in,  0, (size_t)N_NODES * 4, stream);
  hipMemsetAsync(AGG1,  0, (size_t)N_NODES * H_FEATS * 4, stream);
  hipMemsetAsync(out,   0, (size_t)N_NODES * N_CLASSES * 4, stream);

  // degree norms
  deg_kernel<<<(N_EDGES + 255) / 256, 256, 0, stream>>>(src, dst, s_out, s_in, N_EDGES);
  invsqrt_kernel<<<(N_NODES + 255) / 256, 256, 0, stream>>>(s_out, s_in, N_NODES);

  // pack weights into pair-major float2 (one-time, tiny; stays hot in L2)
  {
    int total1 = (IN_FEATS / 2) * H_FEATS;
    pack_w_kernel<<<(total1 + 255) / 256, 256, 0, stream>>>(W1, Wp1, IN_FEATS, H_FEATS, 7);
    int total2 = (H_FEATS / 2) * N_CLASSES;
    pack_w_kernel<<<(total2 + 255) / 256, 256, 0, stream>>>(W2, Wp2, H_FEATS, N_CLASSES, 5);
  }

  // ---- layer 1: transform first (256 -> 128), then aggregate narrow rows ----
  {
    const int mtiles = N_NODES / 16;                 // 6250
    const int ntiles = H_FEATS / 16;                 // 8 waves/block = 256 threads
    const size_t ldsBytes = (size_t)16 * (IN_FEATS + 4) * 4;
    wmma_gemm_f32_lds<<<dim3(mtiles), dim3(ntiles * 32), ldsBytes, stream>>>(
        x, Wp1, Y1, IN_FEATS, H_FEATS, 8);
  }
  {
    long total = (long)N_EDGES * (H_FEATS / 4);
    scatter_kernel<<<(int)((total + 255) / 256), 256, 0, stream>>>(
        Y1, src, dst, s_out, AGG1, N_EDGES, H_FEATS, 5);
  }
  {
    long total = (long)N_NODES * H_FEATS;
    post_kernel<<<(int)((total + 255) / 256), 256, 0, stream>>>(
        AGG1, s_in, b1, N_NODES, 7, 1);
  }

  // ---- layer 2: transform (128 -> 32), aggregate into d_out ----
  {
    const int mtiles = N_NODES / 16;                 // 6250
    const int ntiles = N_CLASSES / 16;               // 2 waves/block = 64 threads
    const size_t ldsBytes = (size_t)16 * (H_FEATS + 4) * 4;
    wmma_gemm_f32_lds<<<dim3(mtiles), dim3(ntiles * 32), ldsBytes, stream>>>(
        AGG1, Wp2, Y2, H_FEATS, N_CLASSES, 7);
  }
  {
    long total = (long)N_EDGES * (N_CLASSES / 4);
    scatter_kernel<<<(int)((total + 255) / 256), 256, 0, stream>>>(
        Y2, src, dst, s_out, out, N_EDGES, N_CLASSES, 3);
  }
  {
    long total = (long)N_NODES * N_CLASSES;
    post_kernel<<<(int)((total + 255) / 256), 256, 0, stream>>>(
        out, s_in, b2, N_NODES, 5, 0);
  }
}